// TargetsLoss_65360812310590
// MI455X (gfx1250) — compile-verified
//
#include <hip/hip_runtime.h>
#include <hip/hip_bf16.h>
#include <math.h>

#define TPB 256
#define NN 2048
#define EPT (NN / TPB)        // 8 elements per thread
#define NWAVES (TPB / 32)     // 8 wave32s per block
#define HDELTA 0.04f
#define LOG_CLAMP_F (-100.0f)

// CDNA5 async global->LDS path (gfx1250). Probe round 1 showed the builtin's
// params are generic pointers to 4 x i32 vectors (no explicit address space,
// non-const), so we cast to that exact shape.
#if __has_builtin(__builtin_amdgcn_global_load_async_to_lds_b128) && \
    __has_builtin(__builtin_amdgcn_s_wait_asynccnt)
#define USE_ASYNC_LDS 1
typedef int v4i_async __attribute__((ext_vector_type(4)));
#endif

__device__ __forceinline__ float wred_sum(float v) {
#pragma unroll
  for (int o = 16; o > 0; o >>= 1) v += __shfl_xor(v, o, 32);
  return v;
}
__device__ __forceinline__ float wred_max(float v) {
#pragma unroll
  for (int o = 16; o > 0; o >>= 1) v = fmaxf(v, __shfl_xor(v, o, 32));
  return v;
}
__device__ __forceinline__ float huber1(float x) {
  float ax = fabsf(x);
  return (ax <= HDELTA) ? 0.5f * x * x : HDELTA * (ax - 0.5f * HDELTA);
}

__global__ __launch_bounds__(TPB) void targets_loss_rows(
    const float* __restrict__ anchors, const float* __restrict__ offsets,
    const float* __restrict__ conf, const float* __restrict__ gt,
    float* __restrict__ ws, int Brows)
{
  __shared__ float sh_conf[NN];      // one confidence row, staged via async DMA
  __shared__ float s_f[NWAVES];
  __shared__ int   s_i[NWAVES];
  __shared__ float s_bcast[2];

  const int b    = blockIdx.x;
  const int t    = threadIdx.x;
  const int lane = t & 31;
  const int wave = t >> 5;

  const float* __restrict__ arow = anchors + (size_t)b * (NN * 2);
  const float* __restrict__ crow = conf    + (size_t)b * NN;
  const float gx = gt[2 * b + 0];
  const float gy = gt[2 * b + 1];

  // ---- kick off the confidence-row copy into LDS (overlaps the anchor pass)
#ifdef USE_ASYNC_LDS
#pragma unroll
  for (int j = 0; j < EPT / 4; ++j) {               // b128 = 4 floats per lane
    const int g = t + j * TPB;
    __builtin_amdgcn_global_load_async_to_lds_b128(
        (v4i_async*)(crow + 4 * g),
        (v4i_async*)&sh_conf[4 * g],
        /*offset=*/0, /*cpol=*/0);
  }
#else
#pragma unroll
  for (int j = 0; j < EPT / 4; ++j) {
    const int g = t + j * TPB;
    *(float4*)&sh_conf[4 * g] = *(const float4*)(crow + 4 * g);
  }
#endif

  // ---- pass 1: streamed argmin of squared distance (sqrt is monotone)
  float best_d = 3.4e38f;
  int   best_i = NN;
  const float4* __restrict__ a4 = (const float4*)arow;   // 2 anchors / float4
#pragma unroll
  for (int k = 0; k < EPT / 2; ++k) {
    const int p = t + k * TPB;                // anchor-pair index
    float4 v = a4[p];
    const float dx0 = v.x - gx, dy0 = v.y - gy;
    const float dx1 = v.z - gx, dy1 = v.w - gy;
    float d0 = dx0 * dx0 + dy0 * dy0;
    const float d1 = dx1 * dx1 + dy1 * dy1;
    int i0 = 2 * p;
    if (d1 < d0) { d0 = d1; i0 = 2 * p + 1; }   // strict <: first-min wins
    if (d0 < best_d) { best_d = d0; best_i = i0; }
  }
  // warp argmin (tie -> smaller index, matching jnp.argmin first occurrence)
#pragma unroll
  for (int o = 16; o > 0; o >>= 1) {
    const float od = __shfl_xor(best_d, o, 32);
    const int   oi = __shfl_xor(best_i, o, 32);
    if (od < best_d || (od == best_d && oi < best_i)) { best_d = od; best_i = oi; }
  }
  if (lane == 0) { s_f[wave] = best_d; s_i[wave] = best_i; }

  // ---- async copy must be complete before any wave reads sh_conf
#ifdef USE_ASYNC_LDS
  __builtin_amdgcn_s_wait_asynccnt(0);
#endif
  __syncthreads();

  if (t == 0) {
    float bd = s_f[0]; int bi = s_i[0];
    for (int w = 1; w < NWAVES; ++w)
      if (s_f[w] < bd || (s_f[w] == bd && s_i[w] < bi)) { bd = s_f[w]; bi = s_i[w]; }
    s_i[0] = bi;
  }
  __syncthreads();
  const int ci = s_i[0];

  // ---- pass 2a: row max of confidences (from LDS, b128 reads)
  float m = -3.4e38f;
#pragma unroll
  for (int j = 0; j < EPT / 4; ++j) {
    const float4 c = *(const float4*)&sh_conf[4 * (t + j * TPB)];
    m = fmaxf(fmaxf(fmaxf(m, c.x), fmaxf(c.y, c.z)), c.w);
  }
  m = wred_max(m);
  if (lane == 0) s_f[wave] = m;
  __syncthreads();
  if (t == 0) {
    float mm = s_f[0];
    for (int w = 1; w < NWAVES; ++w) mm = fmaxf(mm, s_f[w]);
    s_bcast[0] = mm;
  }
  __syncthreads();
  const float M = s_bcast[0];

  // ---- pass 2b: sum of exp(c - M)
  float se = 0.f;
#pragma unroll
  for (int j = 0; j < EPT / 4; ++j) {
    const float4 c = *(const float4*)&sh_conf[4 * (t + j * TPB)];
    se += expf(c.x - M) + expf(c.y - M) + expf(c.z - M) + expf(c.w - M);
  }
  se = wred_sum(se);
  if (lane == 0) s_f[wave] = se;
  __syncthreads();
  if (t == 0) {
    float ss = 0.f;
    for (int w = 0; w < NWAVES; ++w) ss += s_f[w];
    s_bcast[1] = ss;
  }
  __syncthreads();
  const float S    = s_bcast[1];
  const float invS = 1.0f / S;
  const float logS = logf(S);

  // ---- pass 2c: sum of clamped log1p(-p)
  float sl = 0.f;
#pragma unroll
  for (int j = 0; j < EPT / 4; ++j) {
    const float4 c = *(const float4*)&sh_conf[4 * (t + j * TPB)];
    sl += fmaxf(log1pf(-expf(c.x - M) * invS), LOG_CLAMP_F);
    sl += fmaxf(log1pf(-expf(c.y - M) * invS), LOG_CLAMP_F);
    sl += fmaxf(log1pf(-expf(c.z - M) * invS), LOG_CLAMP_F);
    sl += fmaxf(log1pf(-expf(c.w - M) * invS), LOG_CLAMP_F);
  }
  sl = wred_sum(sl);
  if (lane == 0) s_f[wave] = sl;
  __syncthreads();

  if (t == 0) {
    float sum_l1 = 0.f;
    for (int w = 0; w < NWAVES; ++w) sum_l1 += s_f[w];
    const float c_ci = sh_conf[ci];
    const float p_ci = expf(c_ci - M) * invS;
    const float lp   = fmaxf(c_ci - M - logS, LOG_CLAMP_F);     // log p[ci]
    const float l1   = fmaxf(log1pf(-p_ci), LOG_CLAMP_F);       // log(1-p[ci])
    const float ce_row = -(lp + (sum_l1 - l1));

    // Huber on the gathered closest anchor/offset (2 floats each)
    const float ax = arow[2 * ci], ay = arow[2 * ci + 1];
    const float* __restrict__ orow = offsets + (size_t)b * (NN * 2);
    const float ox = orow[2 * ci], oy = orow[2 * ci + 1];
    const float rx = ox - (gx - ax);
    const float ry = oy - (gy - ay);

    ws[b]         = ce_row;
    ws[Brows + b] = huber1(rx) + huber1(ry);
  }
}

__global__ __launch_bounds__(TPB) void targets_loss_finalize(
    const float* __restrict__ ws, float* __restrict__ out, int Brows)
{
  __shared__ float s_ce[NWAVES], s_hu[NWAVES];
  const int t = threadIdx.x, lane = t & 31, wave = t >> 5;
  float ce = 0.f, hu = 0.f;
  for (int i = t; i < Brows; i += TPB) { ce += ws[i]; hu += ws[Brows + i]; }
  ce = wred_sum(ce);
  hu = wred_sum(hu);
  if (lane == 0) { s_ce[wave] = ce; s_hu[wave] = hu; }
  __syncthreads();
  if (t == 0) {
    float tc = 0.f, th = 0.f;
    for (int w = 0; w < NWAVES; ++w) { tc += s_ce[w]; th += s_hu[w]; }
    out[0] = tc + th;   // total_loss
    out[1] = tc;        // ce_loss
    out[2] = th;        // huber_loss
  }
}

extern "C" void kernel_launch(void* const* d_in, const int* in_sizes, int n_in,
                              void* d_out, int out_size, void* d_ws, size_t ws_size,
                              hipStream_t stream) {
  (void)n_in; (void)out_size; (void)ws_size;
  const float* anchors = (const float*)d_in[0];   // (B, N, 2)
  const float* offsets = (const float*)d_in[1];   // (B, N, 2)
  const float* conf    = (const float*)d_in[2];   // (B, N)
  const float* gt      = (const float*)d_in[3];   // (B, 2)
  float* ws  = (float*)d_ws;                      // 2*B floats of scratch
  float* out = (float*)d_out;                     // 3 floats

  const int Brows = in_sizes[3] / 2;              // 8192

  targets_loss_rows<<<dim3(Brows), dim3(TPB), 0, stream>>>(
      anchors, offsets, conf, gt, ws, Brows);
  targets_loss_finalize<<<dim3(1), dim3(TPB), 0, stream>>>(ws, out, Brows);
}